// MultiIndexSelect_41661182771290
// MI455X (gfx1250) — compile-verified
//
#include <hip/hip_runtime.h>

// MultiIndexSelect: out[to_s[i]] = mat_s[from_s[i]] for s in {0,1,2}
// Pure bandwidth problem (~630 MB, 0 FLOPs -> ~27us @ 23.3 TB/s).
// CDNA5 strategy: per-lane async gather global->LDS (B128, ASYNCcnt-tracked),
// 8 row-pairs in flight per wave issued as an unrolled burst, one
// s_wait_asynccnt 0, then an unrolled burst of async scatter stores.
// Data never touches VGPRs -> low register pressure, deep MLP for the gather.
// Fallback (if async builtins unavailable): direct B128 copy + global_prefetch.

typedef float v4f __attribute__((ext_vector_type(4)));
typedef int   v4i __attribute__((ext_vector_type(4)));

#define ROW_D   64            // floats per row
#define ROWB    256           // bytes per row
#define BLOCK   256           // 8 waves (wave32)
#define WAVES   (BLOCK / 32)
#define KDEPTH  8             // async row-pairs in flight per wave
#define SLOTB   512           // bytes per wave slot: 32 lanes x 16B = 2 rows

#if defined(__HIP_DEVICE_COMPILE__) && \
    __has_builtin(__builtin_amdgcn_global_load_async_to_lds_b128) && \
    __has_builtin(__builtin_amdgcn_global_store_async_from_lds_b128)
#define USE_ASYNC 1
#else
#define USE_ASYNC 0
#endif

// Builtin parameter types (from clang diagnostics): int4* in global AS / LDS AS.
typedef __attribute__((address_space(1))) v4i* gptr_v4i;
typedef __attribute__((address_space(3))) v4i* lptr_v4i;
#define AS1(p) ((gptr_v4i)(p))
#define AS3(p) ((lptr_v4i)(p))

__device__ __forceinline__ void wait_async0() {
#if defined(__HIP_DEVICE_COMPILE__)
#if __has_builtin(__builtin_amdgcn_s_wait_asynccnt)
    __builtin_amdgcn_s_wait_asynccnt(0);
#else
    asm volatile("s_wait_asynccnt 0x0" ::: "memory");
#endif
#endif
}

__global__ __launch_bounds__(BLOCK) void MultiIndexSelect_kernel(
    const float* __restrict__ mat0, const float* __restrict__ mat1,
    const float* __restrict__ mat2,
    const int* __restrict__ from0, const int* __restrict__ from1,
    const int* __restrict__ from2,
    const int* __restrict__ to0, const int* __restrict__ to1,
    const int* __restrict__ to2,
    float* __restrict__ out, int L, long long nrows)
{
#if USE_ASYNC
    // --- Async global->LDS->global pipeline (gfx1250 ASYNCcnt path) ---
    __shared__ char smem[WAVES * KDEPTH * SLOTB];   // 32 KB / block

    const int lane   = threadIdx.x & 31;
    const int wave   = threadIdx.x >> 5;
    const int lane16 = lane & 15;                    // float4 slot within a row
    const int half   = lane >> 4;                    // which row of the pair

    const long long gwave  = (long long)blockIdx.x * WAVES + wave;
    const long long nwaves = (long long)gridDim.x * WAVES;
    const long long npairs = (nrows + 1) >> 1;

    char* slot0 = smem + (size_t)wave * (KDEPTH * SLOTB) + (size_t)lane * 16;

    for (long long p = gwave; p < npairs; p += (long long)KDEPTH * nwaves) {
        float* dsts[KDEPTH];

        // Phase 1: unrolled burst of KDEPTH per-lane async gathers into LDS.
#pragma unroll
        for (int k = 0; k < KDEPTH; ++k) {
            long long q   = p + (long long)k * nwaves;
            long long row = 2 * q + half;
            float* d = nullptr;
            if (q < npairs && row < nrows) {
                int s = (int)(row >= (long long)L) + (int)(row >= 2LL * (long long)L);
                int i = (int)(row - (long long)s * (long long)L);
                const float* mat = (s == 0) ? mat0 : ((s == 1) ? mat1 : mat2);
                const int*   fr  = (s == 0) ? from0 : ((s == 1) ? from1 : from2);
                const int*   to  = (s == 0) ? to0   : ((s == 1) ? to1   : to2);
                const float* gsrc = mat + (size_t)fr[i] * ROW_D + (size_t)lane16 * 4;
                d = out + (size_t)to[i] * ROW_D + (size_t)lane16 * 4;
                __builtin_amdgcn_global_load_async_to_lds_b128(
                    AS1(gsrc), AS3(slot0 + k * SLOTB), 0, 0);
            }
            dsts[k] = d;
        }

        wait_async0();   // all gathers landed in LDS (ASYNCcnt == 0)

        // Phase 2: unrolled burst of async scatters LDS -> out (write-once rows).
#pragma unroll
        for (int k = 0; k < KDEPTH; ++k) {
            if (dsts[k]) {
                __builtin_amdgcn_global_store_async_from_lds_b128(
                    AS1(dsts[k]), AS3(slot0 + k * SLOTB), 0, 0);
            }
        }

        wait_async0();   // stores have read LDS; slots reusable next batch
    }
#else
    // --- Fallback: direct vectorized copy with software prefetch ---
    const int lane16 = threadIdx.x & 15;
    const long long tid     = (long long)blockIdx.x * blockDim.x + threadIdx.x;
    const long long group   = tid / 16;
    const long long ngroups = ((long long)gridDim.x * blockDim.x) / 16;

    for (long long j = group; j < nrows; j += ngroups) {
        int s = (int)(j >= (long long)L) + (int)(j >= 2LL * (long long)L);
        int i = (int)(j - (long long)s * (long long)L);
        const float* mat; const int* fr; const int* to;
        if (s == 0)      { mat = mat0; fr = from0; to = to0; }
        else if (s == 1) { mat = mat1; fr = from1; to = to1; }
        else             { mat = mat2; fr = from2; to = to2; }

        int srow = fr[i];
        int drow = to[i];

        long long jn = j + ngroups;
        if (jn < nrows) {
            int sn = (int)(jn >= (long long)L) + (int)(jn >= 2LL * (long long)L);
            int in = (int)(jn - (long long)sn * (long long)L);
            const float* matn = (sn == 0) ? mat0 : ((sn == 1) ? mat1 : mat2);
            const int*   frn  = (sn == 0) ? from0 : ((sn == 1) ? from1 : from2);
            __builtin_prefetch(matn + (size_t)frn[in] * ROW_D + (size_t)lane16 * 4, 0, 1);
        }

        const v4f* src = (const v4f*)(mat + (size_t)srow * ROW_D) + lane16;
        v4f*       dst = (v4f*)(out + (size_t)drow * ROW_D) + lane16;
        v4f v = *src;
        __builtin_nontemporal_store(v, dst);
    }
#endif
}

extern "C" void kernel_launch(void* const* d_in, const int* in_sizes, int n_in,
                              void* d_out, int out_size, void* d_ws, size_t ws_size,
                              hipStream_t stream) {
    const float* mat0  = (const float*)d_in[0];
    const float* mat1  = (const float*)d_in[1];
    const float* mat2  = (const float*)d_in[2];
    const int*   from0 = (const int*)d_in[3];
    const int*   from1 = (const int*)d_in[4];
    const int*   from2 = (const int*)d_in[5];
    const int*   to0   = (const int*)d_in[6];
    const int*   to1   = (const int*)d_in[7];
    const int*   to2   = (const int*)d_in[8];
    float* out = (float*)d_out;

    int       L     = in_sizes[3];                    // 400000
    long long nrows = (long long)out_size / ROW_D;    // 3*L = 1.2M

    // One KDEPTH-batch of row-pairs per wave: 600000 pairs / (8 waves * 8 slots)
    // = 9375 blocks exactly -> every guard folds to true, no tail iterations.
    long long npairs = (nrows + 1) / 2;
    long long blocks = (npairs + (long long)KDEPTH * WAVES - 1) / ((long long)KDEPTH * WAVES);
    if (blocks < 1) blocks = 1;
    if (blocks > 2147483647LL) blocks = 2147483647LL;

    MultiIndexSelect_kernel<<<(int)blocks, BLOCK, 0, stream>>>(
        mat0, mat1, mat2, from0, from1, from2, to0, to1, to2, out, L, nrows);
}